// BatchedBjorckOrthogonalization_55576876810676
// MI455X (gfx1250) — compile-verified
//
#include <hip/hip_runtime.h>
#include <hip/hip_bf16.h>

typedef float v2f __attribute__((ext_vector_type(2)));
typedef float v8f __attribute__((ext_vector_type(8)));

#define NB    1024
#define BATCH 64
#define TILE  128
#define KB    32
#define NKB   (NB / KB)
#define ITERS 13
#define APAD  (KB + 4)     // update-kernel A row pitch: 36 floats = 144B (16B-aligned)
#define BPAD  (TILE + 4)   // k-major tile row pitch: 132 floats = 528B (16B-aligned)

// LDS arena for the gram kernel: staging buffers and the transpose buffer
// occupy the same bytes (phase-separated by barriers).
// 2 arrays * 2 buffers * KB*BPAD floats = 67584 B == TILE*BPAD floats.
#define GRAM_SMEM_BYTES (2 * 2 * KB * BPAD * 4)
static_assert(GRAM_SMEM_BYTES == TILE * BPAD * 4, "transpose buffer must fit staging arena");

// Low 32 bits of a flat LDS-aperture address = byte offset into LDS (ISA 10.2).
__device__ __forceinline__ unsigned lds_off(const void* p) {
    return (unsigned)(size_t)p;
}

// Async global->LDS 16-byte copy (GV mode), tracked by ASYNCcnt.
__device__ __forceinline__ void async_copy_b128(unsigned lds_dst, const float* gsrc) {
    asm volatile("global_load_async_to_lds_b128 %0, %1, off"
                 :: "v"(lds_dst), "v"(gsrc) : "memory");
}

__device__ __forceinline__ void wait_async_all() {
    asm volatile("s_wait_asynccnt 0x0" ::: "memory");
}

// -------------------------------------------------------------------------
// Kernel 1: G[b] = w[b]^T @ w[b]      C[m,n] = sum_k w[k,m] * w[k,n]
// Symmetric: only tiles with n0 <= m0 are computed; off-diagonal tiles are
// mirrored to G[n][m] via an LDS-staged transpose (coalesced stores).
// 256 threads / 8 waves; 128x128 WG tile; 64x32 per wave; double-buffered LDS.
// -------------------------------------------------------------------------
__global__ __launch_bounds__(256) void bjorck_gram_kernel(
    const float* __restrict__ w, float* __restrict__ g)
{
    const int m0 = blockIdx.y * TILE;
    const int n0 = blockIdx.x * TILE;
    if (n0 > m0) return;               // uniform early-out: mirrored tile

    __shared__ __align__(16) char smem[GRAM_SMEM_BYTES];
    float (*As)[KB][BPAD] = (float (*)[KB][BPAD])(smem);                      // [2][KB][BPAD]
    float (*Bs)[KB][BPAD] = (float (*)[KB][BPAD])(smem + 2 * KB * BPAD * 4);  // [2][KB][BPAD]
    float (*Ts)[BPAD]     = (float (*)[BPAD])(smem);                          // [TILE][BPAD]

    const int b  = blockIdx.z;
    const float* wb = w + (size_t)b * NB * NB;
    float*       gb = g + (size_t)b * NB * NB;

    const int t    = threadIdx.x;
    const int lane = t & 31;
    const int wid  = t >> 5;
    const int half = lane >> 4;
    const int ml   = lane & 15;
    const int wr   = wid >> 2;    // wave m offset = wr*64
    const int wc   = wid & 3;     // wave n offset = wc*32

    const int lrow = t >> 5;        // 0..7
    const int lcol = (t & 31) * 4;  // 0..124 step 4

    v8f acc[4][2];
    #pragma unroll
    for (int mi = 0; mi < 4; ++mi)
        #pragma unroll
        for (int ni = 0; ni < 2; ++ni)
            #pragma unroll
            for (int r = 0; r < 8; ++r)
                acc[mi][ni][r] = 0.0f;

    auto prefetch = [&](int buf, int k0) {
        #pragma unroll
        for (int i = 0; i < KB / 8; ++i) {
            const int kk = lrow + i * 8;
            async_copy_b128(lds_off(&As[buf][kk][lcol]),
                            wb + (size_t)(k0 + kk) * NB + m0 + lcol);
            async_copy_b128(lds_off(&Bs[buf][kk][lcol]),
                            wb + (size_t)(k0 + kk) * NB + n0 + lcol);
        }
    };

    prefetch(0, 0);

    for (int kb = 0; kb < NKB; ++kb) {
        const int cur = kb & 1;
        wait_async_all();        // our async writes into buffer `cur` are done
        __syncthreads();         // everyone's are done; `1-cur` free to overwrite

        if (kb + 1 < NKB) prefetch(1 - cur, (kb + 1) * KB);

        #pragma unroll
        for (int ks = 0; ks < KB; ks += 4) {
            v2f af[4], bf[2];
            #pragma unroll
            for (int mi = 0; mi < 4; ++mi) {
                af[mi].x = As[cur][ks + 2 * half + 0][wr * 64 + mi * 16 + ml];
                af[mi].y = As[cur][ks + 2 * half + 1][wr * 64 + mi * 16 + ml];
            }
            #pragma unroll
            for (int ni = 0; ni < 2; ++ni) {
                bf[ni].x = Bs[cur][ks + 2 * half + 0][wc * 32 + ni * 16 + ml];
                bf[ni].y = Bs[cur][ks + 2 * half + 1][wc * 32 + ni * 16 + ml];
            }
            #pragma unroll
            for (int mi = 0; mi < 4; ++mi)
                #pragma unroll
                for (int ni = 0; ni < 2; ++ni)
                    acc[mi][ni] = __builtin_amdgcn_wmma_f32_16x16x4_f32(
                        false, af[mi], false, bf[ni], (short)0, acc[mi][ni],
                        false, false);
        }
    }

    // Direct (coalesced) store of the computed tile: G[m][n]
    #pragma unroll
    for (int mi = 0; mi < 4; ++mi)
        #pragma unroll
        for (int ni = 0; ni < 2; ++ni)
            #pragma unroll
            for (int r = 0; r < 8; ++r) {
                const int m = m0 + wr * 64 + mi * 16 + r + 8 * half;
                const int n = n0 + wc * 32 + ni * 16 + ml;
                gb[(size_t)m * NB + n] = acc[mi][ni][r];
            }

    // Mirror off-diagonal tile to G[n][m] via LDS transpose (coalesced stores)
    if (m0 != n0) {
        __syncthreads();   // all waves finished reading As/Bs (last compute done)
        #pragma unroll
        for (int mi = 0; mi < 4; ++mi)
            #pragma unroll
            for (int ni = 0; ni < 2; ++ni)
                #pragma unroll
                for (int r = 0; r < 8; ++r) {
                    const int ml_loc = wr * 64 + mi * 16 + r + 8 * half;
                    const int nl_loc = wc * 32 + ni * 16 + ml;
                    Ts[nl_loc][ml_loc] = acc[mi][ni][r];
                }
        __syncthreads();
        const int trow = t >> 5;        // 0..7
        const int tcol = (t & 31) * 4;
        #pragma unroll
        for (int i = 0; i < TILE / 8; ++i) {
            const int r = trow + i * 8;
            const float4 v = *(const float4*)&Ts[r][tcol];
            *(float4*)(gb + (size_t)(n0 + r) * NB + m0 + tcol) = v;
        }
    }
}

// -------------------------------------------------------------------------
// Kernel 2: out[b] = 1.5*w[b] - 0.5 * (w[b] @ G[b])
// -------------------------------------------------------------------------
__global__ __launch_bounds__(256) void bjorck_update_kernel(
    const float* __restrict__ w, const float* __restrict__ g,
    float* __restrict__ out)
{
    __shared__ float As[2][TILE][APAD];  // As[buf][m][kk] = w[m0+m][k0+kk]
    __shared__ float Bs[2][KB][BPAD];    // Bs[buf][kk][n] = g[k0+kk][n0+n]

    const int b  = blockIdx.z;
    const int m0 = blockIdx.y * TILE;
    const int n0 = blockIdx.x * TILE;
    const float* wb = w   + (size_t)b * NB * NB;
    const float* gb = g   + (size_t)b * NB * NB;
    float*       ob = out + (size_t)b * NB * NB;

    const int t    = threadIdx.x;
    const int lane = t & 31;
    const int wid  = t >> 5;
    const int half = lane >> 4;
    const int ml   = lane & 15;
    const int wr   = wid >> 2;
    const int wc   = wid & 3;

    const int brow = t >> 5;        // B-stage rows 0..7
    const int bcol = (t & 31) * 4;
    const int arow = t >> 3;        // A-stage rows 0..31
    const int acol = (t & 7) * 4;   // 0..28 step 4 (16B-aligned in LDS: 144B pitch)

    v8f acc[4][2];
    #pragma unroll
    for (int mi = 0; mi < 4; ++mi)
        #pragma unroll
        for (int ni = 0; ni < 2; ++ni)
            #pragma unroll
            for (int r = 0; r < 8; ++r)
                acc[mi][ni][r] = 0.0f;

    auto prefetch = [&](int buf, int k0) {
        #pragma unroll
        for (int i = 0; i < TILE / 32; ++i) {
            const int m = arow + i * 32;
            async_copy_b128(lds_off(&As[buf][m][acol]),
                            wb + (size_t)(m0 + m) * NB + k0 + acol);
        }
        #pragma unroll
        for (int i = 0; i < KB / 8; ++i) {
            const int kk = brow + i * 8;
            async_copy_b128(lds_off(&Bs[buf][kk][bcol]),
                            gb + (size_t)(k0 + kk) * NB + n0 + bcol);
        }
    };

    prefetch(0, 0);

    for (int kb = 0; kb < NKB; ++kb) {
        const int cur = kb & 1;
        wait_async_all();
        __syncthreads();

        if (kb + 1 < NKB) prefetch(1 - cur, (kb + 1) * KB);

        #pragma unroll
        for (int ks = 0; ks < KB; ks += 4) {
            v2f af[4], bf[2];
            #pragma unroll
            for (int mi = 0; mi < 4; ++mi) {
                af[mi].x = As[cur][wr * 64 + mi * 16 + ml][ks + 2 * half + 0];
                af[mi].y = As[cur][wr * 64 + mi * 16 + ml][ks + 2 * half + 1];
            }
            #pragma unroll
            for (int ni = 0; ni < 2; ++ni) {
                bf[ni].x = Bs[cur][ks + 2 * half + 0][wc * 32 + ni * 16 + ml];
                bf[ni].y = Bs[cur][ks + 2 * half + 1][wc * 32 + ni * 16 + ml];
            }
            #pragma unroll
            for (int mi = 0; mi < 4; ++mi)
                #pragma unroll
                for (int ni = 0; ni < 2; ++ni)
                    acc[mi][ni] = __builtin_amdgcn_wmma_f32_16x16x4_f32(
                        false, af[mi], false, bf[ni], (short)0, acc[mi][ni],
                        false, false);
        }
    }

    // Epilogue: out = 1.5*w - 0.5*(w@G)
    #pragma unroll
    for (int mi = 0; mi < 4; ++mi)
        #pragma unroll
        for (int ni = 0; ni < 2; ++ni)
            #pragma unroll
            for (int r = 0; r < 8; ++r) {
                const int m = m0 + wr * 64 + mi * 16 + r + 8 * half;
                const int n = n0 + wc * 32 + ni * 16 + ml;
                const size_t idx = (size_t)m * NB + n;
                ob[idx] = 1.5f * wb[idx] - 0.5f * acc[mi][ni][r];
            }
}

// -------------------------------------------------------------------------
// Driver: 13 Bjorck steps.
//   d_ws[0 .. 256MB)      : G scratch
//   d_ws[256MB .. 512MB)  : w ping buffer
//   d_out                 : w pong buffer (final iteration lands here)
// -------------------------------------------------------------------------
extern "C" void kernel_launch(void* const* d_in, const int* in_sizes, int n_in,
                              void* d_out, int out_size, void* d_ws, size_t ws_size,
                              hipStream_t stream)
{
    (void)in_sizes; (void)n_in; (void)out_size; (void)ws_size;

    const float* w_in = (const float*)d_in[0];
    float* out  = (float*)d_out;
    float* gbuf = (float*)d_ws;
    float* wtmp = (float*)d_ws + (size_t)BATCH * NB * NB;

    const dim3 grid(NB / TILE, NB / TILE, BATCH);
    const dim3 block(256);

    const float* src = w_in;
    for (int it = 0; it < ITERS; ++it) {
        float* dst = ((it & 1) == 0) ? out : wtmp;   // it==12 (last) -> d_out
        bjorck_gram_kernel<<<grid, block, 0, stream>>>(src, gbuf);
        bjorck_update_kernel<<<grid, block, 0, stream>>>(src, gbuf, dst);
        src = dst;
    }
}